// heatsolve_7971459301424
// MI455X (gfx1250) — compile-verified
//
#include <hip/hip_runtime.h>
#include <hip/hip_bf16.h>

// Heat equation, 1024x1024, 100 frames stored. One init kernel + 99 step
// kernels on `stream`. Step kernel: LDS-staged 34x66 halo tile per block,
// per-wave 16x16 output tile computed as two banded matmuls on the WMMA pipe:
//   lap = D(16x20) * Uext(20x16)  +  Uext2(16x20) * E(20x16)
// using chained V_WMMA_F32_16X16X4_F32 (K=4, 10 WMMAs per tile). D and E are
// the same 20x16 second-difference band table, staged once per block in LDS.

#define NX 1024
#define TSTEPS 100
#define LDS_S 66            // LDS row stride in floats (34 rows x 66 cols)
#define LDS_ROWS 34

typedef __attribute__((ext_vector_type(2))) float v2f;
typedef __attribute__((ext_vector_type(8))) float v8f;

// gamma scale = DT / DX^2, computed in double then rounded to f32 (matches jnp)
constexpr double DXd = 6.0 / (NX - 1);
constexpr double DTd = 10.0 / (TSTEPS - 1);
constexpr float GAMMA_C = (float)(DTd / (DXd * DXd));

// Tridiagonal band weight: second-difference operator.
// stencil_w(k, m) = +1 at k==m, -2 at k==m+1, +1 at k==m+2, else 0.
// (Rows k=18,19 of the 20x16 table are all zero -> tail-chunk products vanish.)
__device__ __forceinline__ float stencil_w(int k, int m) {
    float v = 0.0f;
    if (k == m)     v += 1.0f;
    if (k == m + 1) v -= 2.0f;
    if (k == m + 2) v += 1.0f;
    return v;
}

// Frame 0: copy u0 with boundary forced to zero. float4 per thread.
__global__ __launch_bounds__(256) void heat_init(const float* __restrict__ u0,
                                                 float* __restrict__ out0) {
    int idx = blockIdx.x * blockDim.x + threadIdx.x;   // 0 .. 1024*256-1
    int i  = idx >> 8;            // row (256 float4 per 1024-col row)
    int j4 = (idx & 255) << 2;    // starting column of this float4
    float4 v = reinterpret_cast<const float4*>(u0)[idx];
    if (i == 0 || i == NX - 1) { v.x = 0.f; v.y = 0.f; v.z = 0.f; v.w = 0.f; }
    if (j4 == 0)      v.x = 0.f;
    if (j4 == NX - 4) v.w = 0.f;
    reinterpret_cast<float4*>(out0)[idx] = v;
}

// One Euler step: dst = src + gamma .* Lap(src), zero boundary.
// Grid: (16, 32) blocks of 256 threads (8 wave32 waves). Block covers
// 32 rows x 64 cols (2x4 wave tiles of 16x16). Block region is aligned to
// the 64x64 gamma bands (NX/PARAM_GRID == 64), so gamma is uniform per block.
__global__ __launch_bounds__(256) void heat_step(const float* __restrict__ src,
                                                 float* __restrict__ dst,
                                                 const float* __restrict__ alpha) {
    __shared__ float tile[LDS_ROWS * LDS_S];   // u halo tile
    __shared__ float wtab[20 * 16];            // band-weight table, rows 18,19 = 0

    const int tid  = threadIdx.x;
    const int lane = tid & 31;
    const int wid  = tid >> 5;          // 0..7
    const int row0 = blockIdx.y * 32;   // first output row of this block
    const int col0 = blockIdx.x * 64;   // first output col of this block

    // ---- Stage u halo (34x66) row-per-wave: scalar-friendly row addressing.
    // Out-of-grid halo is zero-filled; it only feeds boundary outputs, which
    // are forced to zero anyway (matches the reference's clamped boundary).
    for (int r = wid; r < LDS_ROWS; r += 8) {
        int gr = row0 - 1 + r;
        const float* srow = src + (long)gr * NX + (col0 - 1);
        bool rowok = (unsigned)gr < (unsigned)NX;
        for (int c = lane; c < LDS_S; c += 32) {
            int gc = col0 - 1 + c;
            float v = 0.0f;
            if (rowok && (unsigned)gc < (unsigned)NX) v = srow[c];
            tile[r * LDS_S + c] = v;
        }
    }
    // ---- Stage band-weight table (one pass, 320 entries).
    if (tid < 20 * 16) {
        wtab[tid] = stencil_w(tid >> 4, tid & 15);
    }
    __syncthreads();

    // ---- Per-wave 16x16 tile via WMMA f32 16x16x4 (wave32).
    const int hf  = lane >> 4;       // lane half (K split across halves)
    const int ln  = lane & 15;       // M (for A) / N (for B,C,D)
    const int tr  = wid >> 2;        // tile row in block: 0..1
    const int tc  = wid & 3;         // tile col in block: 0..3
    const int lr0 = 1 + tr * 16;     // LDS row of output row 0 of this tile
    const int lc0 = 1 + tc * 16;     // LDS col of output col 0 of this tile

    // Band weights: used as A of the vertical matmul (D[m][k], lane m = ln)
    // and as B of the horizontal matmul (E[k][n], lane n = ln) — identical.
    // A layout (f32 16x4): lane = M, VGPR j = K = 4kk + 2*hf + j.
    // B layout (f32 4x16): VGPR j = row K = 4kk + 2*hf + j, lanes = N.
    v2f w[5];
#pragma unroll
    for (int kk = 0; kk < 5; ++kk) {
        int K0 = kk * 4 + 2 * hf;
        w[kk].x = wtab[K0 * 16 + ln];
        w[kk].y = wtab[K0 * 16 + 16 + ln];
    }

    // Vertical B operands: rows of u. Tail chunk (kk=4, hf=1 -> K=18,19) is
    // index-clamped; the matching weight rows are zero, so no guard needed.
    v2f ub[5];
#pragma unroll
    for (int kk = 0; kk < 5; ++kk) {
        int r0, r1;
        if (kk < 4) { r0 = kk * 4 + 2 * hf; r1 = r0 + 1; }
        else        { r0 = 16 + hf;         r1 = 17;     }
        ub[kk].x = tile[(lr0 - 1 + r0) * LDS_S + lc0 + ln];
        ub[kk].y = tile[(lr0 - 1 + r1) * LDS_S + lc0 + ln];
    }

    // Horizontal A operands: cols of u, same clamped tail.
    v2f ua[5];
#pragma unroll
    for (int kk = 0; kk < 5; ++kk) {
        int c0, c1;
        if (kk < 4) { c0 = kk * 4 + 2 * hf; c1 = c0 + 1; }
        else        { c0 = 16 + hf;         c1 = 17;     }
        ua[kk].x = tile[(lr0 + ln) * LDS_S + lc0 - 1 + c0];
        ua[kk].y = tile[(lr0 + ln) * LDS_S + lc0 - 1 + c1];
    }

    // 10 chained WMMAs accumulate the full Laplacian.
    v8f acc = {};
#pragma unroll
    for (int kk = 0; kk < 5; ++kk)
        acc = __builtin_amdgcn_wmma_f32_16x16x4_f32(
            false, w[kk], false, ub[kk], (short)0, acc, false, false);
#pragma unroll
    for (int kk = 0; kk < 5; ++kk)
        acc = __builtin_amdgcn_wmma_f32_16x16x4_f32(
            false, ua[kk], false, w[kk], (short)0, acc, false, false);

    // ---- Epilogue: u_next = u + gamma * lap, zero boundary.
    // Block is aligned to the 64x64 nearest-upsample bands -> uniform gamma.
    const float g = alpha[(row0 >> 6) * 16 + (col0 >> 6)] * GAMMA_C;

    const int gj   = col0 + tc * 16 + ln;
    const bool jb  = (gj == 0) || (gj == NX - 1);
    // C/D layout: VGPR r holds M = r (lanes 0-15) and M = r + 8 (lanes 16-31).
#pragma unroll
    for (int r = 0; r < 8; ++r) {
        int m  = r + 8 * hf;
        int gi = row0 + tr * 16 + m;
        float uc = tile[(lr0 + m) * LDS_S + lc0 + ln];
        float v  = uc + g * acc[r];
        if (jb || gi == 0 || gi == NX - 1) v = 0.0f;
        dst[(long)gi * NX + gj] = v;
    }
}

extern "C" void kernel_launch(void* const* d_in, const int* in_sizes, int n_in,
                              void* d_out, int out_size, void* d_ws, size_t ws_size,
                              hipStream_t stream) {
    const float* u0    = (const float*)d_in[0];   // [1024, 1024] f32
    const float* alpha = (const float*)d_in[1];   // [16, 16] f32
    float* out = (float*)d_out;                   // [100, 1024, 1024] f32

    const size_t frame = (size_t)NX * NX;

    // Frame 0: u0 with zeroed boundary.
    heat_init<<<(NX * NX / 4) / 256, 256, 0, stream>>>(u0, out);

    // Frames 1..99: sequential Euler steps (each reads the previous frame).
    dim3 grid(NX / 64, NX / 32);   // (16, 32)
    for (int k = 1; k < TSTEPS; ++k) {
        heat_step<<<grid, 256, 0, stream>>>(out + (size_t)(k - 1) * frame,
                                            out + (size_t)k * frame,
                                            alpha);
    }
}